// GCN_47674136985850
// MI455X (gfx1250) — compile-verified
//
#include <hip/hip_runtime.h>
#include <math.h>

// ---------------------------------------------------------------------------
// Problem constants (from reference)
// ---------------------------------------------------------------------------
#define N_NODES 8192
#define N_EDGES 262144
#define FEAT    512
#define HID     512
#define NOUT    64

typedef float v2f __attribute__((ext_vector_type(2)));
typedef float v8f __attribute__((ext_vector_type(8)));

// ---------------------------------------------------------------------------
// Native FP32 WMMA: D(16x16,f32) = A(16x4,f32) x B(4x16,f32) + C
// A frag (2 VGPRs/lane): lane<16 -> row=lane,     {K=k,   K=k+1}
//                        lane>=16 -> row=lane-16, {K=k+2, K=k+3}
// C/D (8 VGPRs/lane):    lane<16 -> rows 0..7, col=lane
//                        lane>=16 -> rows 8..15, col=lane-16
// ---------------------------------------------------------------------------
__device__ __forceinline__ v8f wmma_f32x4(v2f a, v2f b, v8f c) {
  return __builtin_amdgcn_wmma_f32_16x16x4_f32(
      /*neg_a=*/false, a, /*neg_b=*/false, b,
      /*c_mod=*/(short)0, c, /*reuse_a=*/false, /*reuse_b=*/false);
}

// ---------------------------------------------------------------------------
// LDS-staged WMMA GEMM:  C[M,N] = A[M,K] x op(B) (+ epilogue)
//   BT=false : B is row-major [K,N]          (transposed while staging to LDS)
//   BT=true  : B is row-major [N,K], used as B^T (contiguous staging)
//   EPI=0: plain store   EPI=1: + bias aux[col]
//   EPI=2: cosine normalize v/(aux[row]*aux[col]), 1e-8 guard, zero diagonal
//
// Block tile BM x BN, K-chunk BK=32 staged in LDS (row stride 34 floats:
// 8B-aligned for ds_load_b64, conflict-free for the 16-lane fragment reads).
// Each wave computes a 32x64 tile = 2x4 WMMA tiles (64 accumulator VGPRs).
// blockDim.x = (BM/32)*(BN/64)*32
// ---------------------------------------------------------------------------
template <int BM, int BN, bool BT, int EPI>
__global__ void wmma_gemm(const float* __restrict__ A,
                          const float* __restrict__ B,
                          float* __restrict__ C,
                          const float* __restrict__ aux,
                          int M, int N, int K) {
  constexpr int BK    = 32;
  constexpr int LDW   = BK + 2;   // 34-float padded LDS row stride
  constexpr int WROWS = BM / 32;
  constexpr int WCOLS = BN / 64;
  constexpr int NTHR  = WROWS * WCOLS * 32;

  __shared__ float smem[(BM + BN) * LDW];
  float* As = smem;
  float* Bs = smem + BM * LDW;

  const int tid  = threadIdx.x;
  const int lane = tid & 31;
  const int w    = tid >> 5;
  const int wm   = w / WCOLS;
  const int wn   = w % WCOLS;

  const int mBase = blockIdx.y * BM;
  const int nBase = blockIdx.x * BN;

  const int frow   = lane & 15;         // fragment row/col within 16
  const int kk_off = (lane >> 4) << 1;  // lanes 16-31 take K+2,K+3
  const int m_off  = wm * 32;
  const int n_off  = wn * 64;

  v8f acc[2][4];
#pragma unroll
  for (int i = 0; i < 2; ++i)
#pragma unroll
    for (int j = 0; j < 4; ++j)
#pragma unroll
      for (int r = 0; r < 8; ++r) acc[i][j][r] = 0.0f;

  for (int k0 = 0; k0 < K; k0 += BK) {
    // ---- stage A tile: As[r][c] = A[mBase+r][k0+c] (contiguous b64 stores)
    for (int idx = tid; idx < BM * (BK / 4); idx += NTHR) {
      const int r  = idx >> 3;          // BK/4 == 8
      const int c4 = (idx & 7) << 2;
      const float4 v = *(const float4*)(A + (size_t)(mBase + r) * K + k0 + c4);
      float* d = &As[r * LDW + c4];
      ((float2*)d)[0] = make_float2(v.x, v.y);
      ((float2*)d)[1] = make_float2(v.z, v.w);
    }
    // ---- stage B tile as Bs[n][c] = op(B)[k0+c][nBase+n]
    if (BT) {
      for (int idx = tid; idx < BN * (BK / 4); idx += NTHR) {
        const int r  = idx >> 3;
        const int c4 = (idx & 7) << 2;
        const float4 v = *(const float4*)(B + (size_t)(nBase + r) * K + k0 + c4);
        float* d = &Bs[r * LDW + c4];
        ((float2*)d)[0] = make_float2(v.x, v.y);
        ((float2*)d)[1] = make_float2(v.z, v.w);
      }
    } else {
      for (int idx = tid; idx < BK * (BN / 4); idx += NTHR) {
        const int kk = idx / (BN / 4);
        const int n4 = (idx % (BN / 4)) << 2;
        const float4 v = *(const float4*)(B + (size_t)(k0 + kk) * N + nBase + n4);
        Bs[(n4 + 0) * LDW + kk] = v.x;
        Bs[(n4 + 1) * LDW + kk] = v.y;
        Bs[(n4 + 2) * LDW + kk] = v.z;
        Bs[(n4 + 3) * LDW + kk] = v.w;
      }
    }
    __syncthreads();

    // ---- 8 K-steps x 8 WMMAs, all operands from LDS
#pragma unroll
    for (int ks = 0; ks < BK; ks += 4) {
      const int kk = ks + kk_off;
      const v2f a0 = *(const v2f*)&As[(m_off + frow) * LDW + kk];
      const v2f a1 = *(const v2f*)&As[(m_off + 16 + frow) * LDW + kk];
      v2f b[4];
#pragma unroll
      for (int j = 0; j < 4; ++j)
        b[j] = *(const v2f*)&Bs[(n_off + j * 16 + frow) * LDW + kk];
#pragma unroll
      for (int j = 0; j < 4; ++j) {
        acc[0][j] = wmma_f32x4(a0, b[j], acc[0][j]);
        acc[1][j] = wmma_f32x4(a1, b[j], acc[1][j]);
      }
    }
    __syncthreads();
  }

  // ---- epilogue
#pragma unroll
  for (int i = 0; i < 2; ++i)
#pragma unroll
    for (int j = 0; j < 4; ++j) {
      const int rbase = mBase + m_off + i * 16 + ((lane >> 4) << 3);
      const int col   = nBase + n_off + j * 16 + frow;
#pragma unroll
      for (int r = 0; r < 8; ++r) {
        const int row = rbase + r;
        float v = acc[i][j][r];
        if (EPI == 1) v += aux[col];
        if (EPI == 2) {
          float denom = aux[row] * aux[col];
          denom = (denom == 0.0f) ? 1e-8f : denom;
          v = (row == col) ? 0.0f : v / denom;
        }
        C[(size_t)row * N + col] = v;
      }
    }
}

// ---------------------------------------------------------------------------
// Elementwise / graph kernels
// ---------------------------------------------------------------------------
__global__ void fill_f32(float* __restrict__ p, float v, size_t n) {
  size_t i = (size_t)blockIdx.x * blockDim.x + threadIdx.x;
  if (i < n) p[i] = v;
}

__global__ void deg_accum(const int* __restrict__ dst, float* __restrict__ deg,
                          int E) {
  int i = blockIdx.x * blockDim.x + threadIdx.x;
  if (i < E) atomicAdd(&deg[dst[i]], 1.0f);
}

__global__ void deg_rsqrt(float* __restrict__ p, int n) {
  int i = blockIdx.x * blockDim.x + threadIdx.x;
  if (i < n) p[i] = rsqrtf(p[i]);  // deg >= 1 (self-loop), no zero guard needed
}

// out[dst[e], f..f+3] += h[src[e], f..f+3] * dinv[src]*dinv[dst]
__global__ void scatter_edges(const float* __restrict__ h,
                              float* __restrict__ out,
                              const int* __restrict__ src,
                              const int* __restrict__ dst,
                              const float* __restrict__ dinv,
                              int E, int F) {
  const int per = F >> 2;  // float4 chunks per edge
  int idx = blockIdx.x * blockDim.x + threadIdx.x;
  if (idx >= E * per) return;
  const int e = idx / per;
  const int f = (idx - e * per) << 2;
  const int s = src[e];
  const int d = dst[e];
  const float nrm = dinv[s] * dinv[d];
  const float4 v = *(const float4*)(h + (size_t)s * F + f);
  float* o = out + (size_t)d * F + f;
  atomicAdd(o + 0, v.x * nrm);
  atomicAdd(o + 1, v.y * nrm);
  atomicAdd(o + 2, v.z * nrm);
  atomicAdd(o + 3, v.w * nrm);
}

// agg = relu(agg + g*dinv[row]^2 + bias[f])  -- self-loop term fused, no atomics
__global__ void bias_relu_self(float* __restrict__ agg,
                               const float* __restrict__ g,
                               const float* __restrict__ dinv,
                               const float* __restrict__ bias,
                               int Nn, int F) {
  size_t idx = (size_t)blockIdx.x * blockDim.x + threadIdx.x;
  if (idx >= (size_t)Nn * F) return;
  const int row = (int)(idx / F);
  const int f   = (int)(idx - (size_t)row * F);
  const float di = dinv[row];
  float v = agg[idx] + g[idx] * di * di + bias[f];
  agg[idx] = v > 0.0f ? v : 0.0f;
}

// one wave per row: norms[r] = sqrt(sum_f h[r,f]^2)
__global__ void row_norms(const float* __restrict__ h,
                          float* __restrict__ norms, int F) {
  const int row  = blockIdx.x * (blockDim.x >> 5) + (threadIdx.x >> 5);
  const int lane = threadIdx.x & 31;
  const float* p = h + (size_t)row * F;
  float s = 0.0f;
  for (int f = lane; f < F; f += 32) {
    float v = p[f];
    s += v * v;
  }
#pragma unroll
  for (int off = 16; off > 0; off >>= 1) s += __shfl_xor(s, off, 32);
  if (lane == 0) norms[row] = sqrtf(s);
}

// ---------------------------------------------------------------------------
// Launch
// ---------------------------------------------------------------------------
extern "C" void kernel_launch(void* const* d_in, const int* in_sizes, int n_in,
                              void* d_out, int out_size, void* d_ws,
                              size_t ws_size, hipStream_t stream) {
  const float* x   = (const float*)d_in[0];
  const int*   ei  = (const int*)d_in[1];
  const float* W1  = (const float*)d_in[2];
  const float* b1  = (const float*)d_in[3];
  const float* W2  = (const float*)d_in[4];
  const float* b2  = (const float*)d_in[5];
  const float* fcW = (const float*)d_in[6];
  const float* fcb = (const float*)d_in[7];

  const int* src = ei;            // edge_index row 0
  const int* dst = ei + N_EDGES;  // edge_index row 1

  float* logits = (float*)d_out;                                 // [8192,64]
  float* xdis   = (float*)d_out + (size_t)N_NODES * NOUT;        // [8192,8192]

  float* dinv  = (float*)d_ws;                                   // [N]
  float* norms = dinv + N_NODES;                                 // [N]
  float* bufG  = norms + N_NODES;                                // [N,HID] gemm out
  float* bufH  = bufG + (size_t)N_NODES * HID;                   // [N,HID] agg/feat

  const size_t NH = (size_t)N_NODES * HID;

  // ---- symmetric normalization: dinv = (1 + indeg)^-1/2 ----
  fill_f32<<<(N_NODES + 255) / 256, 256, 0, stream>>>(dinv, 1.0f, N_NODES);
  deg_accum<<<(N_EDGES + 255) / 256, 256, 0, stream>>>(dst, dinv, N_EDGES);
  deg_rsqrt<<<(N_NODES + 255) / 256, 256, 0, stream>>>(dinv, N_NODES);

  // ---- layer 1: bufG = x @ W1 ; aggregate ; relu ----
  wmma_gemm<128, 128, false, 0>
      <<<dim3(HID / 128, N_NODES / 128), 256, 0, stream>>>(
          x, W1, bufG, nullptr, N_NODES, HID, FEAT);
  fill_f32<<<(unsigned)((NH + 255) / 256), 256, 0, stream>>>(bufH, 0.0f, NH);
  scatter_edges<<<(N_EDGES * (HID / 4)) / 256, 256, 0, stream>>>(
      bufG, bufH, src, dst, dinv, N_EDGES, HID);
  bias_relu_self<<<(unsigned)((NH + 255) / 256), 256, 0, stream>>>(
      bufH, bufG, dinv, b1, N_NODES, HID);

  // ---- layer 2: bufG = h1 @ W2 ; aggregate ; relu -> x_hid in bufH ----
  wmma_gemm<128, 128, false, 0>
      <<<dim3(HID / 128, N_NODES / 128), 256, 0, stream>>>(
          bufH, W2, bufG, nullptr, N_NODES, HID, HID);
  fill_f32<<<(unsigned)((NH + 255) / 256), 256, 0, stream>>>(bufH, 0.0f, NH);
  scatter_edges<<<(N_EDGES * (HID / 4)) / 256, 256, 0, stream>>>(
      bufG, bufH, src, dst, dinv, N_EDGES, HID);
  bias_relu_self<<<(unsigned)((NH + 255) / 256), 256, 0, stream>>>(
      bufH, bufG, dinv, b2, N_NODES, HID);

  // ---- logits = x_hid @ fcW + fcb  (128x64 block tile, 4 waves) ----
  wmma_gemm<128, 64, false, 1>
      <<<dim3(1, N_NODES / 128), 128, 0, stream>>>(
          bufH, fcW, logits, fcb, N_NODES, NOUT, HID);

  // ---- cosine-similarity matrix with fused normalization + zero diag ----
  row_norms<<<N_NODES / 8, 256, 0, stream>>>(bufH, norms, HID);
  wmma_gemm<128, 128, true, 2>
      <<<dim3(N_NODES / 128, N_NODES / 128), 256, 0, stream>>>(
          bufH, bufH, xdis, norms, N_NODES, N_NODES, HID);
}